// Baseline_33079838114572
// MI455X (gfx1250) — compile-verified
//
#include <hip/hip_runtime.h>
#include <hip/hip_bf16.h>
#include <math.h>

typedef __attribute__((ext_vector_type(2))) float v2f;
typedef __attribute__((ext_vector_type(8))) float v8f;

// ---------------------------------------------------------------------------
// Small utility kernels
// ---------------------------------------------------------------------------

__global__ void k_zero_out(float* __restrict__ out) {
  if (threadIdx.x < 2) out[threadIdx.x] = 0.0f;
}

// Concatenate user/item embeddings into x0 and seed the layer accumulator.
__global__ void k_init(const float4* __restrict__ ue, const float4* __restrict__ ie,
                       float4* __restrict__ x, float4* __restrict__ acc,
                       long uquads, long total) {
  long i = (long)blockIdx.x * blockDim.x + threadIdx.x;
  if (i >= total) return;
  float4 v = (i < uquads) ? ue[i] : ie[i - uquads];
  x[i]   = v;
  acc[i] = v;
}

__global__ void k_zero_buf(float4* __restrict__ p, long total) {
  long i = (long)blockIdx.x * blockDim.x + threadIdx.x;
  if (i >= total) return;
  float4 z; z.x = 0.f; z.y = 0.f; z.z = 0.f; z.w = 0.f;
  p[i] = z;
}

__global__ void k_acc_add(float4* __restrict__ acc, const float4* __restrict__ y, long total) {
  long i = (long)blockIdx.x * blockDim.x + threadIdx.x;
  if (i >= total) return;
  float4 a = acc[i];
  float4 b = y[i];
  a.x += b.x; a.y += b.y; a.z += b.z; a.w += b.w;
  acc[i] = a;
}

// ---------------------------------------------------------------------------
// SpMM layer: y[dst] += x[src] * val   (D = 64, 16 lanes/edge, float4/lane)
// Edge arrays are streamed with non-temporal loads so the ~77MB of x/y tables
// stay resident in the 192MB L2 across the random gather/scatter.
// ---------------------------------------------------------------------------

__global__ void k_spmm(const float* __restrict__ x, float* __restrict__ y,
                       const float* __restrict__ vals, const int* __restrict__ src,
                       const int* __restrict__ dst, long E) {
  long t = (long)blockIdx.x * blockDim.x + threadIdx.x;
  long e = t >> 4;           // 16 lanes per edge
  int part = (int)(t & 15);  // which float4 of the 64-float row
  if (e >= E) return;
  int s  = __builtin_nontemporal_load(src + e);
  int d  = __builtin_nontemporal_load(dst + e);
  float v = __builtin_nontemporal_load(vals + e);
  float4 m = ((const float4*)(x + (long)s * 64))[part];  // coalesced 256B/half-wave
  float* yr = y + (long)d * 64 + part * 4;
  atomicAdd(yr + 0, m.x * v);
  atomicAdd(yr + 1, m.y * v);
  atomicAdd(yr + 2, m.z * v);
  atomicAdd(yr + 3, m.w * v);
}

// ---------------------------------------------------------------------------
// BPR scoring epilogue via V_WMMA_F32_16X16X4_F32.
// One wave handles a tile of 16 batch rows: C_pos = U * Pos^T, C_neg = U * Neg^T
// (16 K-steps of 4). Diagonal of C gives the per-row dot products.
//   A layout (16x4 f32): lane L -> row M=L%16; VGPR r -> K = r + (L<16?0:2)
//   B layout (4x16 f32): lane L -> col N=L%16; VGPR r -> K = r + (L<16?0:2)
//   C layout: lane L holds N=L%16; VGPR r holds M = r + (L<16?0:8)
//   => diag element b: lanes 0..7 at r=lane, lanes 24..31 at r=lane-24
// ---------------------------------------------------------------------------

__global__ void k_score(const float* __restrict__ acc,
                        const int* __restrict__ users, const int* __restrict__ pos,
                        const int* __restrict__ neg, int n_users,
                        float* __restrict__ loss_out, int batch) {
  int gt = blockIdx.x * blockDim.x + threadIdx.x;
#if __has_builtin(__builtin_amdgcn_wmma_f32_16x16x4_f32)
  int wave = gt >> 5;
  int lane = gt & 31;
  int m = lane & 15;
  int b = wave * 16 + m;
  if (b >= batch) b = batch - 1;  // clamp instead of return: EXEC must stay all-1s
  const float* urow = acc + (long)users[b] * 64;
  const float* prow = acc + (long)(n_users + pos[b]) * 64;
  const float* nrow = acc + (long)(n_users + neg[b]) * 64;
  int cbase = (lane < 16) ? 0 : 2;
  v8f cp = {};
  v8f cn = {};
#pragma unroll
  for (int kb = 0; kb < 16; ++kb) {
    int col = kb * 4 + cbase;               // even -> 8B aligned
    v2f a  = *(const v2f*)(urow + col);
    v2f bp = *(const v2f*)(prow + col);
    v2f bn = *(const v2f*)(nrow + col);
    cp = __builtin_amdgcn_wmma_f32_16x16x4_f32(false, a, false, bp, (short)0, cp, false, false);
    cn = __builtin_amdgcn_wmma_f32_16x16x4_f32(false, a, false, bn, (short)0, cn, false, false);
  }
  int r = (lane < 8) ? lane : ((lane >= 24) ? (lane - 24) : -1);
  if (r >= 0) {
    int bdiag = wave * 16 + (lane & 15);
    if (bdiag < batch) {
      const float scale = 1.0f / 16.0f;     // (1/4 layer-mean)^2
      float ps = cp[r] * scale;
      float ns = cn[r] * scale;
      float x = ns - ps;
      float sp = fmaxf(x, 0.0f) + log1pf(expf(-fabsf(x)));  // stable softplus
      atomicAdd(loss_out, sp);
    }
  }
#else
  // Fallback VALU path (correct, no WMMA) — histogram will flag this case.
  if (gt < batch) {
    const float* urow = acc + (long)users[gt] * 64;
    const float* prow = acc + (long)(n_users + pos[gt]) * 64;
    const float* nrow = acc + (long)(n_users + neg[gt]) * 64;
    float ps = 0.f, ns = 0.f;
    for (int d = 0; d < 64; ++d) { ps += urow[d] * prow[d]; ns += urow[d] * nrow[d]; }
    float x = (ns - ps) * (1.0f / 16.0f);
    float sp = fmaxf(x, 0.0f) + log1pf(expf(-fabsf(x)));
    atomicAdd(loss_out, sp);
  }
#endif
}

__global__ void k_reg(const float* __restrict__ ue, const float* __restrict__ ie,
                      const int* __restrict__ users, const int* __restrict__ pos,
                      const int* __restrict__ neg, float* __restrict__ reg_out, int batch) {
  int b = blockIdx.x * blockDim.x + threadIdx.x;
  if (b >= batch) return;
  const float* u = ue + (long)users[b] * 64;
  const float* p = ie + (long)pos[b] * 64;
  const float* n = ie + (long)neg[b] * 64;
  float s = 0.f;
#pragma unroll 4
  for (int d = 0; d < 64; ++d) s += u[d] * u[d] + p[d] * p[d] + n[d] * n[d];
  atomicAdd(reg_out, 0.5f * s / (float)batch);
}

// ---------------------------------------------------------------------------
// Host-side launch sequence
// ---------------------------------------------------------------------------

extern "C" void kernel_launch(void* const* d_in, const int* in_sizes, int n_in,
                              void* d_out, int out_size, void* d_ws, size_t ws_size,
                              hipStream_t stream) {
  const float* user_emb  = (const float*)d_in[0];
  const float* item_emb  = (const float*)d_in[1];
  const float* edge_vals = (const float*)d_in[2];
  const int*   edge_src  = (const int*)d_in[3];
  const int*   edge_dst  = (const int*)d_in[4];
  const int*   users     = (const int*)d_in[5];
  const int*   pos       = (const int*)d_in[6];
  const int*   neg       = (const int*)d_in[7];

  const long n_users = in_sizes[0] / 64;
  const long n_items = in_sizes[1] / 64;
  const long E       = in_sizes[2];
  const int  batch   = in_sizes[5];
  const long N  = n_users + n_items;
  const long ND = N * 64;
  const long quads = ND / 4;

  float* buf0 = (float*)d_ws;          // ping
  float* buf1 = buf0 + ND;             // pong
  float* accb = buf1 + ND;             // layer-sum accumulator
  float* out  = (float*)d_out;         // [0]=loss, [1]=reg_loss

  const int TB = 256;
  const long qb = (quads + TB - 1) / TB;

  k_zero_out<<<1, 32, 0, stream>>>(out);
  k_init<<<(unsigned)qb, TB, 0, stream>>>((const float4*)user_emb, (const float4*)item_emb,
                                          (float4*)buf0, (float4*)accb,
                                          n_users * 16, quads);

  float* xc = buf0;
  float* xn = buf1;
  for (int layer = 0; layer < 3; ++layer) {
    k_zero_buf<<<(unsigned)qb, TB, 0, stream>>>((float4*)xn, quads);
    long threads = E * 16;
    long eb = (threads + TB - 1) / TB;
    k_spmm<<<(unsigned)eb, TB, 0, stream>>>(xc, xn, edge_vals, edge_src, edge_dst, E);
    k_acc_add<<<(unsigned)qb, TB, 0, stream>>>((float4*)accb, (const float4*)xn, quads);
    float* t = xc; xc = xn; xn = t;
  }

  // one wave per 16 batch rows; batch=4096 -> 256 full waves, EXEC all-1s
  long sthreads = (long)((batch + 15) / 16) * 32;
  k_score<<<(unsigned)((sthreads + TB - 1) / TB), TB, 0, stream>>>(
      accb, users, pos, neg, (int)n_users, out, batch);
  k_reg<<<(unsigned)((batch + TB - 1) / TB), TB, 0, stream>>>(
      user_emb, item_emb, users, pos, neg, out + 1, batch);
}